// IIRConv2d_4982162063860
// MI455X (gfx1250) — compile-verified
//
#include <hip/hip_runtime.h>
#include <cstdint>

// Problem dims (from reference)
constexpr int Bc = 4, Cc = 32, Hc = 256, Wc = 1024;
constexpr int K1 = 5, K2 = 3;

// Tiling
constexpr int ROWS  = 128;       // rows per block == threads per block (4 waves)
constexpr int WROWS = 32;        // rows DMA'd per wave (one slab per wave)
constexpr int CW    = 64;        // W-chunk per stage
constexpr int TILE0 = 72;        // TDM tile_dim0: dwords per staged row
constexpr int LROW  = 81;        // padded LDS row stride (72 + 72/8), conflict-free
constexpr int OG    = CW / 4;    // 16 b128 groups per output row
constexpr int OWP   = 68;        // out-tile LDS stride (16B-aligned, 2-way worst)
constexpr int NCHUNK = Wc / CW;  // 16

typedef uint32_t v4u __attribute__((ext_vector_type(4)));
typedef int      v8i __attribute__((ext_vector_type(8)));
typedef int      v4i __attribute__((ext_vector_type(4)));

static __device__ __forceinline__ uint32_t lds_lo32(const void* p) {
  // Low 32 bits of a generic pointer into LDS == wave-relative LDS byte offset.
  return (uint32_t)(uintptr_t)p;
}

// One 2D DMA: WROWS rows x TILE0 dwords, row stride Wc, global -> LDS,
// LDS padded 1 dword per 8 (row stride LROW). Tracked by TENSORcnt.
static __device__ __forceinline__ void tdm_load_2d(uint32_t lds_addr,
                                                   const float* gaddr,
                                                   uint32_t valid_cols) {
  const uint64_t ga = (uint64_t)(uintptr_t)gaddr;
  v4u g0;
  g0.x = 1u;                                            // count=1 (valid user D#)
  g0.y = lds_addr;                                      // lds_addr
  g0.z = (uint32_t)ga;                                  // global_addr[31:0]
  g0.w = (uint32_t)((ga >> 32) & 0x01FFFFFFu) | (2u << 30);  // ga[56:32]|type=2

  const uint32_t flags = (2u << 16)   // data_size = 4 bytes
                       | (1u << 20)   // pad_enable
                       | (2u << 22)   // pad_interval: every 8 dwords
                       | (0u << 25);  // pad_amount: 1 dword
  const uint32_t td0 = valid_cols;    // OOB cols (>= td0) read back as zero
  const uint32_t td1 = (uint32_t)WROWS;
  v8i g1;
  g1[0] = (int)flags;
  g1[1] = (int)((td0 & 0xFFFFu) << 16);                      // tensor_dim0[15:0]
  g1[2] = (int)((td0 >> 16) | ((td1 & 0xFFFFu) << 16));      // td0 hi | td1 lo
  g1[3] = (int)((td1 >> 16) | ((uint32_t)TILE0 << 16));      // td1 hi | tile_dim0
  g1[4] = (int)((uint32_t)WROWS);                            // tile_dim1 (tile_dim2=0)
  g1[5] = (int)Wc;                                           // tensor_dim0_stride lo
  g1[6] = 0;
  g1[7] = 0;
  const v4i gz4 = {0, 0, 0, 0};                              // 2D: groups 2/3 inert
  const v8i gz8 = {0, 0, 0, 0, 0, 0, 0, 0};
#if __has_builtin(__builtin_amdgcn_tensor_load_to_lds)
  // clang-23 / therock-10.0 form: 6 args (g0, g1, g2, g3, extra, cpol)
  __builtin_amdgcn_tensor_load_to_lds(g0, g1, gz4, gz4, gz8, 0);
#else
  asm volatile("tensor_load_to_lds %0, %1, %2, %3"
               :: "s"(g0), "s"(g1), "s"(gz4), "s"(gz4) : "memory");
#endif
}

static __device__ __forceinline__ void wait_tensor0() {
#if __has_builtin(__builtin_amdgcn_s_wait_tensorcnt)
  __builtin_amdgcn_s_wait_tensorcnt(0);
#else
  asm volatile("s_wait_tensorcnt 0x0" ::: "memory");
#endif
}

static __device__ __forceinline__ void async_store_b128(float* gaddr,
                                                        uint32_t lds_addr) {
  asm volatile("global_store_async_from_lds_b128 %0, %1, off th:TH_STORE_NT"
               :: "v"(gaddr), "v"(lds_addr)
               : "memory");
}

static __device__ __forceinline__ void wait_async0() {
  asm volatile("s_wait_asynccnt 0" ::: "memory");
}

__global__ __launch_bounds__(ROWS)
void iir_conv_kernel(const float* __restrict__ x,
                     const float* __restrict__ w1,   // (C,1,K1)
                     const float* __restrict__ w2,   // (C,K2)
                     float* __restrict__ out) {
  __shared__ __align__(16) float inT[2][ROWS * LROW]; // double-buffered DMA tiles
  __shared__ __align__(16) float outT[ROWS * OWP];    // output staging tile

  const int tid     = threadIdx.x;
  // Wave id, pinned uniform into an SGPR (tensor descriptors want scalar args;
  // every wave executes the TDM op -> no EXEC-masking hazard, and each wave's
  // TENSORcnt tracks exactly its own 32-row slab).
  const int wid     = __builtin_amdgcn_readfirstlane(tid >> 5);
  const int rowBase = blockIdx.x * ROWS;        // flat row = (b*C + c)*H + h
  const int c       = (rowBase / Hc) % Cc;      // whole block is one channel

  const float h0 = w1[c * K1 + 0], h1 = w1[c * K1 + 1], h2 = w1[c * K1 + 2],
              h3 = w1[c * K1 + 3], h4 = w1[c * K1 + 4];
  const float a0 = w2[c * K2 + 0], a1 = w2[c * K2 + 1], a2 = w2[c * K2 + 2];

  const float* xrow0 = x   + (size_t)rowBase * Wc;
  float*       orow0 = out + (size_t)rowBase * Wc;

  float s1 = 0.f, s2 = 0.f, s3 = 0.f;   // IIR state: out[w-1], out[w-2], out[w-3]

  // Chunk k tile starts at global col (k==0 ? 0 : k*CW-4).
  // Each wave DMAs rows [wid*32, wid*32+32) of the block's 128-row slab.
  auto issue_tdm = [&](int buf, int k) {
    const int ts = (k == 0) ? 0 : (k * CW - 4);
    tdm_load_2d(lds_lo32(&inT[buf][(size_t)wid * WROWS * LROW]),
                xrow0 + (size_t)wid * WROWS * Wc + ts,
                (uint32_t)(Wc - ts));
  };

  issue_tdm(0, 0);

  for (int k = 0; k < NCHUNK; ++k) {
    const int buf = k & 1;

    wait_tensor0();             // this wave's chunk-k slab landed in LDS
    wait_async0();              // this wave's chunk-(k-1) output stores done
    __syncthreads();            // publish all slabs to all waves

    if (k + 1 < NCHUNK) issue_tdm(buf ^ 1, k + 1);   // overlap next DMA

    // ---- fused FIR(5) + IIR(3), one thread per row, sliding window ----
    // padded LDS col for tile col c: c + c/8 ; tile col of w: w - ts
    const float* rin = &inT[buf][tid * LROW];
    auto pc = [](int col) { return col + (col >> 3); };
    const int cb = (k == 0) ? 0 : 4;        // tile col of w0
    float xm2, xm1, xc0, xp1;
    if (k == 0) {                           // x[-2], x[-1] are zero-padding
      xm2 = 0.f;            xm1 = 0.f;
      xc0 = rin[pc(0)];     xp1 = rin[pc(1)];
    } else {
      xm2 = rin[pc(cb - 2)]; xm1 = rin[pc(cb - 1)];
      xc0 = rin[pc(cb)];     xp1 = rin[pc(cb + 1)];
    }
    const int w0 = k * CW;
#pragma unroll 8
    for (int i = 0; i < CW; ++i) {
      const float xp2 = rin[pc(cb + i + 2)];
      const float y = h0 * xm2 + h1 * xm1 + h2 * xc0 + h3 * xp1 + h4 * xp2;
      xm2 = xm1; xm1 = xc0; xc0 = xp1; xp1 = xp2;

      const int w = w0 + i;
      const float o = ((w == 0) | (w >= Wc - 2))
                          ? y
                          : y - (a2 * s1 + a1 * s2 + a0 * s3);
      s3 = s2; s2 = s1; s1 = o;
      outT[tid * OWP + i] = o;
    }
    __syncthreads();            // outT complete; inT(buf) reads done

    // ---- cooperative async B128 store-from-LDS (coalesced, NT) ----
    for (int e = tid; e < ROWS * OG; e += ROWS) {   // 16 iters
      const int r = e / OG, g4 = e % OG;
      async_store_b128(orow0 + (size_t)r * Wc + (w0 + g4 * 4),
                       lds_lo32(&outT[r * OWP + g4 * 4]));
    }
  }

  wait_async0();                // drain stores (S_ENDPGM also waits all counters)
}

extern "C" void kernel_launch(void* const* d_in, const int* in_sizes, int n_in,
                              void* d_out, int out_size, void* d_ws, size_t ws_size,
                              hipStream_t stream) {
  const float* x  = (const float*)d_in[0];   // (B,C,H,W) f32
  const float* w1 = (const float*)d_in[1];   // (C,1,K1)  f32
  const float* w2 = (const float*)d_in[2];   // (C,K2)    f32
  float* out      = (float*)d_out;           // (B,C,H,W) f32

  const int totalRows = Bc * Cc * Hc;        // 32768
  dim3 grid(totalRows / ROWS);               // 256 blocks
  dim3 block(ROWS);                          // 128 threads = 4 waves (wave32)
  hipLaunchKernelGGL(iir_conv_kernel, grid, block, 0, stream, x, w1, w2, out);
}